// GradModule_26792005992600
// MI455X (gfx1250) — compile-verified
//
#include <hip/hip_runtime.h>

typedef float v2f __attribute__((ext_vector_type(2)));
typedef float v8f __attribute__((ext_vector_type(8)));

#if defined(__HIP_DEVICE_COMPILE__)
#if !__has_builtin(__builtin_amdgcn_wmma_f32_16x16x4_f32)
#error "missing __builtin_amdgcn_wmma_f32_16x16x4_f32 on gfx1250 device pass"
#endif
#endif

#define HDIM 256
#define CDIM 32
#define NCOL 96            // C*3 output columns
#define NT   (NCOL / 16)   // 6 column tiles
#define U_STRIDE 260       // padded LDS row stride (dwords) -> conflict-free b64 loads
#define KSTEPS (HDIM / 4)  // 64 WMMA k-steps
#define TILES_PER_WAVE 2   // 32 nodes per wave (M blocking, reuses B loads)
#define WAVES_PER_BLOCK 8

// ---------------- Kernel 1: build U^T[col=c*3+k][h] = W2[h,c] * W1[k,h] -----
__global__ void build_ut_kernel(const float* __restrict__ W1,
                                const float* __restrict__ W2,
                                float* __restrict__ Ut) {
  int i = blockIdx.x * blockDim.x + threadIdx.x;
  if (i >= NCOL * HDIM) return;
  int col = i >> 8;          // 0..95
  int h   = i & 255;         // 0..255
  int c   = col / 3;
  int k   = col - 3 * c;
  Ut[col * HDIM + h] = W2[h * CDIM + c] * W1[k * HDIM + h];
}

// sech^2(z) = 1 - tanh^2(z) = 4 e / (1+e)^2,  e = exp(-2|z|)
__device__ __forceinline__ float sech2(float z) {
  float e = __expf(-2.0f * fabsf(z));
  float u = 1.0f + e;
  return (4.0f * e) * __builtin_amdgcn_rcpf(u * u);
}

// ---------------- Kernel 2: out[N,96] = sech2(pos@W1+b1) @ U ----------------
__global__ void __launch_bounds__(WAVES_PER_BLOCK * 32)
jac_kernel(const float* __restrict__ pos,
           const float* __restrict__ W1,
           const float* __restrict__ b1,
           const float* __restrict__ Ut,
           float* __restrict__ out, int N) {
  extern __shared__ float lds[];
  float* sU  = lds;                         // 96 rows x U_STRIDE
  float* sW1 = lds + NCOL * U_STRIDE;       // [3][256]
  float* sB1 = sW1 + 3 * HDIM;              // [256]

  const int tid = threadIdx.x;

  // Cooperative LDS staging (coalesced global reads)
  for (int i = tid; i < NCOL * HDIM; i += blockDim.x) {
    int r = i >> 8, c = i & 255;
    sU[r * U_STRIDE + c] = Ut[i];
  }
  for (int i = tid; i < 3 * HDIM; i += blockDim.x) sW1[i] = W1[i];
  for (int i = tid; i < HDIM;     i += blockDim.x) sB1[i] = b1[i];
  __syncthreads();

  const int lane = tid & 31;
  const int wave = tid >> 5;
  const int m    = lane & 15;              // row within 16-tile / col within 16-tile
  const int koff = (lane >> 4) << 1;       // lanes 0-15 -> K{0,1}, lanes 16-31 -> K{2,3}
  const long tile0 = ((long)blockIdx.x * WAVES_PER_BLOCK + wave) * TILES_PER_WAVE;

  // Per-lane node rows (one per M tile); clamp OOB loads, guard stores later.
  float px[TILES_PER_WAVE], py[TILES_PER_WAVE], pz[TILES_PER_WAVE];
#pragma unroll
  for (int t = 0; t < TILES_PER_WAVE; ++t) {
    long n  = (tile0 + t) * 16 + m;
    long ni = (n < N) ? n : 0;
    px[t] = pos[ni * 3 + 0];
    py[t] = pos[ni * 3 + 1];
    pz[t] = pos[ni * 3 + 2];
  }

  // Hoisted LDS base pointers for this lane
  const float* uRow = sU + m * U_STRIDE;    // + nt*16*U_STRIDE + h0

  v8f acc[TILES_PER_WAVE][NT] = {};         // 2 x 6 x 8 VGPR accumulators

#pragma unroll 2
  for (int kb = 0; kb < KSTEPS; ++kb) {
    const int h0 = (kb << 2) + koff;        // even -> 8B-aligned LDS b64 loads

    // ---- Issue ALL LDS loads first (distinct regs -> one dscnt wait), ----
    // ---- their latency hides under the sech2 VALU/TRANS work below.     ----
    v2f b[NT];
#pragma unroll
    for (int nt = 0; nt < NT; ++nt)
      b[nt] = *(const v2f*)(uRow + nt * 16 * U_STRIDE + h0);

    v2f w0 = *(const v2f*)(sW1 + 0 * HDIM + h0);
    v2f w1 = *(const v2f*)(sW1 + 1 * HDIM + h0);
    v2f w2 = *(const v2f*)(sW1 + 2 * HDIM + h0);
    v2f bb = *(const v2f*)(sB1 + h0);

    // A fragments: s = sech^2(z) computed on the fly
    v2f a[TILES_PER_WAVE];
#pragma unroll
    for (int t = 0; t < TILES_PER_WAVE; ++t) {
      float z0 = fmaf(px[t], w0.x, fmaf(py[t], w1.x, fmaf(pz[t], w2.x, bb.x)));
      float z1 = fmaf(px[t], w0.y, fmaf(py[t], w1.y, fmaf(pz[t], w2.y, bb.y)));
      a[t].x = sech2(z0);
      a[t].y = sech2(z1);
    }

    // ---- 12 back-to-back WMMAs ----
#pragma unroll
    for (int nt = 0; nt < NT; ++nt)
#pragma unroll
      for (int t = 0; t < TILES_PER_WAVE; ++t)
        acc[t][nt] = __builtin_amdgcn_wmma_f32_16x16x4_f32(
            false, a[t], false, b[nt], (short)0, acc[t][nt], false, false);
  }

  // Store D: VGPR r holds row M = r (+8 for upper half-wave), col N = m.
  // Fast path (all 8 rows of this half-wave in range): one base address,
  // 48 stores with immediate offsets -> clauses of global_store_b32.
  const int rowoff = (lane >> 4) << 3;      // 0 or 8
#pragma unroll
  for (int t = 0; t < TILES_PER_WAVE; ++t) {
    const long n0 = (tile0 + t) * 16 + rowoff;
    float* bp = out + n0 * NCOL + m;
    if (n0 + 7 < N) {
#pragma unroll
      for (int r = 0; r < 8; ++r)
#pragma unroll
        for (int nt = 0; nt < NT; ++nt)
          bp[r * NCOL + nt * 16] = acc[t][nt][r];
    } else {
#pragma unroll
      for (int r = 0; r < 8; ++r)
        if (n0 + r < N)
#pragma unroll
          for (int nt = 0; nt < NT; ++nt)
            bp[r * NCOL + nt * 16] = acc[t][nt][r];
    }
  }
}

extern "C" void kernel_launch(void* const* d_in, const int* in_sizes, int n_in,
                              void* d_out, int out_size, void* d_ws, size_t ws_size,
                              hipStream_t stream) {
  const float* pos = (const float*)d_in[0];   // [N,3]
  const float* W1  = (const float*)d_in[1];   // [3,256]
  const float* b1  = (const float*)d_in[2];   // [256]
  const float* W2  = (const float*)d_in[3];   // [256,32]
  float* out = (float*)d_out;                 // [N,32,3] == [N,96]
  float* Ut  = (float*)d_ws;                  // U^T: 96*256 floats (96 KB scratch)

  const int N = in_sizes[0] / 3;

  // Stage 1: tiny fused-weight precompute
  build_ut_kernel<<<(NCOL * HDIM + 255) / 256, 256, 0, stream>>>(W1, W2, Ut);

  // Stage 2: WMMA GEMM  out = sech2(pos@W1+b1) @ U
  const int tiles  = (N + 15) / 16;
  const int waves  = (tiles + TILES_PER_WAVE - 1) / TILES_PER_WAVE;
  const int blocks = (waves + WAVES_PER_BLOCK - 1) / WAVES_PER_BLOCK;
  const size_t shmem = (size_t)(NCOL * U_STRIDE + 3 * HDIM + HDIM) * sizeof(float);

  jac_kernel<<<blocks, WAVES_PER_BLOCK * 32, shmem, stream>>>(pos, W1, b1, Ut, out, N);
}